// WDL_66331474919972
// MI455X (gfx1250) — compile-verified
//
#include <hip/hip_runtime.h>
#include <stdint.h>
#include <math.h>

typedef float v2f __attribute__((ext_vector_type(2)));
typedef float v8f __attribute__((ext_vector_type(8)));

#define B_TOT   16384
#define D       16
#define IN1     80        // 5*D
#define H1      128
#define H2      64
#define CROSS   200000

// LDS layouts (floats):
// weights are stored K-pair interleaved: w[kpair][n][2]  (pair = rows k, k+1)
// pair strides chosen so (stride mod 64) == 32 -> K-halves hit opposite bank halves
#define W1P_STRIDE 288     // 128*2 + 32
#define W2P_STRIDE 160     // 64*2 + 32
#define DX_STRIDE  84      // 80 + 4  (conflict-free A fragments)
#define H1_STRIDE  132     // 128 + 4
#define H2_STRIDE  68      // 64 + 4

#define NWAVES  8          // waves per block (wave32); 16 rows/wave -> 128 rows/block

static constexpr int W1P_FLOATS  = (IN1 / 2) * W1P_STRIDE;   // 40*288 = 11520
static constexpr int W2P_FLOATS  = (H1 / 2) * W2P_STRIDE;    // 64*160 = 10240
static constexpr int WAVE_DX     = 16 * DX_STRIDE;           // 1344 (reused as hid2 16x68 = 1088)
static constexpr int WAVE_H1     = 16 * H1_STRIDE;           // 2112
static constexpr int WAVE_FLOATS = WAVE_DX + WAVE_H1;        // 3456
static constexpr size_t SMEM_BYTES =
    (size_t)(W1P_FLOATS + W2P_FLOATS + NWAVES * WAVE_FLOATS) * sizeof(float); // ~193 KB

__device__ __forceinline__ int64_t hash_pair_dev(int64_t a, int64_t b) {
    const uint64_t K1 = 11400714785074694791ull;
    const uint64_t K2 = 13787848793156543929ull;
    const uint64_t K3 = 10723151780598845931ull;
    int64_t x = (int64_t)((uint64_t)a * K1 + (uint64_t)b);
    x = (int64_t)((uint64_t)(x ^ (x >> 30)) * K2);
    x = (int64_t)((uint64_t)(x ^ (x >> 27)) * K3);
    x = x ^ (x >> 31);
    int64_t h = (x < 0) ? -x : x;        // matches jnp.abs (int64-min edge ignored)
    h %= (int64_t)CROSS;
    if (h < 0) h += CROSS;               // Python floor-mod safety
    return h;
}

__global__ void __launch_bounds__(256)
wdl_fused_kernel(const int* __restrict__ user,
                 const int* __restrict__ codes,       // [B,4]
                 const float* __restrict__ user_emb,  // [2M,16]
                 const float* __restrict__ sem_emb,   // [4,256,16]
                 const float* __restrict__ wide_w,    // [200000,1]
                 const float* __restrict__ w1, const float* __restrict__ b1,
                 const float* __restrict__ w2, const float* __restrict__ b2,
                 const float* __restrict__ w3, const float* __restrict__ b3,
                 float* __restrict__ out)
{
    extern __shared__ float smem[];
    float* w1p = smem;                       // [40][288]  pairs of w1 rows
    float* w2p = smem + W1P_FLOATS;          // [64][160]  pairs of w2 rows

    const int tid  = threadIdx.x;
    const int wave = tid >> 5;
    const int lane = tid & 31;

    float* dx  = w2p + W2P_FLOATS + wave * WAVE_FLOATS; // [16][84]; later reused as hid2 [16][68]
    float* h1s = dx + WAVE_DX;                          // [16][132]

    // ---- stage w1/w2 into LDS as K-pair-interleaved fragments (whole block) ----
    // pair p = rows (2p, 2p+1); element (k,n) lives at wNp[p*STRIDE + 2n + (k&1)]
    #pragma unroll 4
    for (int idx = tid; idx < (IN1 / 2) * H1; idx += 256) {     // 5120 pairs
        const int kp = idx >> 7, n = idx & 127;
        v2f v;
        v.x = w1[(2 * kp)     * H1 + n];
        v.y = w1[(2 * kp + 1) * H1 + n];
        *(v2f*)(w1p + kp * W1P_STRIDE + 2 * n) = v;
    }
    #pragma unroll 4
    for (int idx = tid; idx < (H1 / 2) * H2; idx += 256) {      // 4096 pairs
        const int kp = idx >> 6, n = idx & 63;
        v2f v;
        v.x = w2[(2 * kp)     * H2 + n];
        v.y = w2[(2 * kp + 1) * H2 + n];
        *(v2f*)(w2p + kp * W2P_STRIDE + 2 * n) = v;
    }

    const int row0 = blockIdx.x * (NWAVES * 16) + wave * 16;

    // ---- gather deep features: 16 rows x 80 cols into LDS (per-wave) ----
    for (int idx = lane; idx < 16 * IN1; idx += 32) {   // 1280/32 = 40, no divergence
        const int r   = idx / IN1;
        const int col = idx - r * IN1;
        const int row = row0 + r;
        float v;
        if (col < D) {
            v = user_emb[user[row] * D + col];
        } else {
            const int lev = (col - D) >> 4;
            const int e   = (col - D) & 15;
            int c = codes[row * 4 + lev];
            c = c < 0 ? 0 : (c > 255 ? 255 : c);
            v = sem_emb[((lev << 8) + c) * D + e];
        }
        dx[r * DX_STRIDE + col] = v;
    }

    __syncthreads();   // weights visible block-wide; dx is per-wave (DScnt-ordered)

    const int mrow  = lane & 15;   // M (A/C) or N (B/C) index within 16
    const int khalf = lane >> 4;   // K-half select for A/B fragments

    // Loop-invariant LDS base pointers: all per-lane terms absorbed here so every
    // ds_load/ds_store below uses a pure compile-time immediate offset.
    const float* ap1 = dx  + mrow * DX_STRIDE + 2 * khalf;          // A frags, GEMM1
    const float* ap2 = h1s + mrow * H1_STRIDE + 2 * khalf;          // A frags, GEMM2
    const float* bp1base = w1p + khalf * W1P_STRIDE;                // + 2n per tile
    const float* bp2base = w2p + khalf * W2P_STRIDE;
    float* hp1base = h1s + (khalf * 8) * H1_STRIDE;                 // + n per tile
    float* hp2base = dx  + (khalf * 8) * H2_STRIDE;

    // ---- hoist GEMM1 A fragments into registers (reused across all 8 N-tiles) ----
    v2f a1[IN1 / 4];
    #pragma unroll
    for (int kk = 0; kk < IN1 / 4; ++kk) {
        a1[kk] = *(const v2f*)(ap1 + kk * 4);
    }

    // ---- GEMM1: [16x80] @ [80x128] + b1, ReLU -> h1s ----
    #pragma unroll
    for (int nt = 0; nt < H1 / 16; ++nt) {
        const int n = nt * 16 + mrow;
        const float bias = b1[n];
        const float* bp = bp1base + 2 * n;
        float* hp = hp1base + n;
        v8f c = {bias, bias, bias, bias, bias, bias, bias, bias};
        #pragma unroll
        for (int kk = 0; kk < IN1 / 4; ++kk) {
            // pair (2kk+khalf): byte offset kk*2304 from bp (constant per kk)
            v2f b = *(const v2f*)(bp + kk * (2 * W1P_STRIDE));
            c = __builtin_amdgcn_wmma_f32_16x16x4_f32(false, a1[kk], false, b,
                                                      (short)0, c, false, false);
        }
        #pragma unroll
        for (int r = 0; r < 8; ++r) {
            float v = c[r];
            v = v > 0.f ? v : 0.f;
            hp[r * H1_STRIDE] = v;                 // D(m = r+8*khalf, n)
        }
    }

    // ---- hoist GEMM2 A fragments (reused across all 4 N-tiles) ----
    v2f a2[H1 / 4];
    #pragma unroll
    for (int kk = 0; kk < H1 / 4; ++kk) {
        a2[kk] = *(const v2f*)(ap2 + kk * 4);
    }

    // ---- GEMM2: [16x128] @ [128x64] + b2, ReLU -> hid2 (reuse dx region) ----
    #pragma unroll
    for (int nt = 0; nt < H2 / 16; ++nt) {
        const int n = nt * 16 + mrow;
        const float bias = b2[n];
        const float* bp = bp2base + 2 * n;
        float* hp = hp2base + n;
        v8f c = {bias, bias, bias, bias, bias, bias, bias, bias};
        #pragma unroll
        for (int kk = 0; kk < H1 / 4; ++kk) {
            v2f b = *(const v2f*)(bp + kk * (2 * W2P_STRIDE));   // offset kk*1280 B
            c = __builtin_amdgcn_wmma_f32_16x16x4_f32(false, a2[kk], false, b,
                                                      (short)0, c, false, false);
        }
        #pragma unroll
        for (int r = 0; r < 8; ++r) {
            float v = c[r];
            v = v > 0.f ? v : 0.f;
            hp[r * H2_STRIDE] = v;                 // hid2 tile [16][68]
        }
    }

    // ---- final: w3 dot (float4-vectorized) + wide hash-gather + sigmoid ----
    if (lane < 16) {
        const int row = row0 + lane;
        const int64_t u = (int64_t)user[row];
        float ws = 0.f;
        #pragma unroll
        for (int lev = 0; lev < 4; ++lev) {
            int c = codes[row * 4 + lev];
            c = c < 0 ? 0 : (c > 255 ? 255 : c);
            ws += wide_w[hash_pair_dev(u, (int64_t)c)];
        }
        float acc = b3[0] + ws;
        const float4* hrow = (const float4*)(dx + lane * H2_STRIDE); // 16B-aligned
        const float4* w3v  = (const float4*)w3;
        #pragma unroll
        for (int j = 0; j < H2 / 4; ++j) {
            float4 h = hrow[j];
            float4 w = w3v[j];
            acc += h.x * w.x + h.y * w.y + h.z * w.z + h.w * w.w;
        }
        out[row] = 1.f / (1.f + expf(-acc));
    }
}

extern "C" void kernel_launch(void* const* d_in, const int* in_sizes, int n_in,
                              void* d_out, int out_size, void* d_ws, size_t ws_size,
                              hipStream_t stream) {
    (void)in_sizes; (void)n_in; (void)out_size; (void)d_ws; (void)ws_size;

    const int*   user     = (const int*)  d_in[0];
    const int*   codes    = (const int*)  d_in[1];
    const float* user_emb = (const float*)d_in[2];
    const float* sem_emb  = (const float*)d_in[3];
    const float* wide_w   = (const float*)d_in[4];
    const float* w1       = (const float*)d_in[5];
    const float* b1       = (const float*)d_in[6];
    const float* w2       = (const float*)d_in[7];
    const float* b2       = (const float*)d_in[8];
    const float* w3       = (const float*)d_in[9];
    const float* b3       = (const float*)d_in[10];
    float*       out      = (float*)d_out;

    // allow >64KB dynamic LDS (WGP has 320KB); deterministic, capture-safe
    hipFuncSetAttribute(reinterpret_cast<const void*>(wdl_fused_kernel),
                        hipFuncAttributeMaxDynamicSharedMemorySize,
                        (int)SMEM_BYTES);

    dim3 grid(B_TOT / (NWAVES * 16));  // 128 blocks
    dim3 block(NWAVES * 32);           // 256 threads = 8 wave32
    wdl_fused_kernel<<<grid, block, SMEM_BYTES, stream>>>(
        user, codes, user_emb, sem_emb, wide_w, w1, b1, w2, b2, w3, b3, out);
}